// SoftmaxAttention_86749749444742
// MI455X (gfx1250) — compile-verified
//
#include <hip/hip_runtime.h>

// SoftmaxAttention over labels: out[b,l,d] = sum_s softmax_s(x[b,s,:]·emb[b,l,:]) * x[b,s,d]
// B=16, S=512, D=512, L=8192, f32 end-to-end via V_WMMA_F32_16X16X4_F32.
// CDNA5 paths: v_wmma_f32_16x16x4_f32 + global_load_async_to_lds_b128 (ASYNCcnt).

typedef float v2f __attribute__((ext_vector_type(2)));
typedef float v8f __attribute__((ext_vector_type(8)));
typedef __attribute__((address_space(3))) float lds_f32;

#define NEGV (-100000.0f)

constexpr int kB  = 16;
constexpr int kS  = 512;
constexpr int kD  = 512;
constexpr int kL  = 8192;
constexpr int kLT = 64;     // labels per workgroup
constexpr int kEP = 74;     // score-tile pitch (floats): conflict-free stores & transposed loads
constexpr int kKC = 32;     // K-chunk of the label tile staged in LDS per async copy
constexpr int kLP = 36;     // label-buffer pitch (floats): 144B rows (16B aligned), conflict-free
// LDS: e[kS][kEP], then a region shared (phase-disjoint) by the label staging
// buffer (64*kLP floats) and the softmax scratch (pmax/psum/colmax/colinv).
constexpr int kSmemFloats = kS * kEP + kLT * kLP;   // 40192 floats = 160768 B -> 2 WGs/WGP

__device__ __forceinline__ v8f wmma_f32_16x16x4(v2f a, v2f b, v8f c) {
  // 8 args: (neg_a, A, neg_b, B, c_mod, C, reuse_a, reuse_b)
  return __builtin_amdgcn_wmma_f32_16x16x4_f32(false, a, false, b, (short)0, c,
                                               false, false);
}

__device__ __forceinline__ unsigned lds_off(const float* p) {
  return (unsigned)(unsigned long long)(lds_f32*)p;   // addrspacecast -> LDS byte offset
}

__global__ __launch_bounds__(256)
void SoftmaxAttention_86749749444742_kernel(const float* __restrict__ x,
                                            const unsigned char* __restrict__ mask,
                                            const float* __restrict__ lab,
                                            float* __restrict__ out) {
  extern __shared__ float smem[];
  float* e    = smem;                   // [kS][kEP] score / exp tile
  float* lbuf = smem + kS * kEP;        // [64][kLP] staged label K-chunk (phase 1 only)
  float* pmax   = lbuf;                 // [4][64]  } aliased with lbuf: used only
  float* psum   = pmax + 256;           // [4][64]  } after phase 1 completes
  float* colmax = psum + 256;           // [64]
  float* colinv = colmax + 64;          // [64]

  const int tid  = threadIdx.x;
  const int lane = tid & 31;
  const int wave = tid >> 5;            // 0..7
  const int half = lane >> 4;           // 0/1 (K half of f32 WMMA A/B fragment layout)
  const int l16  = lane & 15;

  const int b  = blockIdx.y;
  const int l0 = blockIdx.x * kLT;

  const float* xb = x   + (size_t)b * kS * kD;
  const float* lb = lab + ((size_t)b * kL + l0) * kD;

  v8f zero8;
#pragma unroll
  for (int i = 0; i < 8; ++i) zero8[i] = 0.0f;

  // ---------------- Phase 1: scores[s0..s0+63][0..63] = x @ labT ----------------
  {
    const int s0 = wave * 64;
    v8f acc[4][4];
#pragma unroll
    for (int mt = 0; mt < 4; ++mt)
#pragma unroll
      for (int nt = 0; nt < 4; ++nt) acc[mt][nt] = zero8;

    const float* arow[4];
#pragma unroll
    for (int t = 0; t < 4; ++t)
      arow[t] = xb + (size_t)(s0 + 16 * t + l16) * kD + 2 * half;   // A: M = s row
    const float* lrow = lbuf + l16 * kLP + 2 * half;                // B frag base in LDS

    for (int kc = 0; kc < kD; kc += kKC) {
      __syncthreads();                  // WAR: previous chunk fully consumed
      // Cooperative async copy: label rows 0..63, cols kc..kc+31 -> lbuf.
      // 64 rows * 128B = 512 x 16B segments; 256 threads x 2.
#pragma unroll
      for (int ss = 0; ss < 2; ++ss) {
        const int seg = tid + 256 * ss;
        const int r = seg >> 3, cs = seg & 7;
        const unsigned loff = lds_off(lbuf + r * kLP + cs * 4);
        const float* g = lb + (size_t)r * kD + kc + cs * 4;
        asm volatile("global_load_async_to_lds_b128 %0, %1, off"
                     :: "v"(loff), "v"((unsigned long long)(uintptr_t)g)
                     : "memory");
      }
      asm volatile("s_wait_asynccnt 0x0" ::: "memory");
      __syncthreads();                  // RAW: chunk visible to all waves

#pragma unroll 1
      for (int kl = 0; kl < kKC; kl += 4) {
        const int k = kc + kl;
        v2f a[4], bf[4];
#pragma unroll
        for (int t = 0; t < 4; ++t) a[t]  = *(const v2f*)(arow[t] + k);
#pragma unroll
        for (int t = 0; t < 4; ++t) bf[t] = *(const v2f*)(lrow + t * (16 * kLP) + kl);
#pragma unroll
        for (int mt = 0; mt < 4; ++mt)
#pragma unroll
          for (int nt = 0; nt < 4; ++nt)
            acc[mt][nt] = wmma_f32_16x16x4(a[mt], bf[nt], acc[mt][nt]);
      }
    }

    __syncthreads();                    // last chunk consumed everywhere before e spill? (e
                                        // region is disjoint from lbuf; barrier keeps phases clean)
    // Spill raw scores to LDS: e[s][l], C-frag layout (VGPR i -> M = i + 8*half).
#pragma unroll
    for (int mt = 0; mt < 4; ++mt)
#pragma unroll
      for (int nt = 0; nt < 4; ++nt) {
        const int c = 16 * nt + l16;
#pragma unroll
        for (int i = 0; i < 8; ++i) {
          const int r = s0 + 16 * mt + i + 8 * half;
          e[r * kEP + c] = acc[mt][nt][i];
        }
      }
  }
  __syncthreads();

  // ---------------- Softmax over s for each label column ----------------
  {
    const int col  = tid & 63;
    const int part = tid >> 6;          // 4 partial scanners per column
    const unsigned char* mrow = mask + (size_t)b * kS;

    float m = NEGV;
    for (int r = part; r < kS; r += 4) {
      const float v = mrow[r] ? e[r * kEP + col] : NEGV;
      m = fmaxf(m, v);
    }
    pmax[part * 64 + col] = m;
    __syncthreads();
    if (part == 0) {
      colmax[col] = fmaxf(fmaxf(pmax[col], pmax[64 + col]),
                          fmaxf(pmax[128 + col], pmax[192 + col]));
    }
    __syncthreads();
    const float cm = colmax[col];
    float s = 0.0f;
    for (int r = part; r < kS; r += 4) {
      const float v = mrow[r] ? __expf(e[r * kEP + col] - cm) : 0.0f;
      e[r * kEP + col] = v;             // store unnormalized exp
      s += v;
    }
    psum[part * 64 + col] = s;
    __syncthreads();
    if (part == 0) {                    // >= S/2 live rows -> sum >= 1, safe
      colinv[col] = 1.0f / (psum[col] + psum[64 + col] +
                            psum[128 + col] + psum[192 + col]);
    }
  }
  __syncthreads();

  // ---------------- Phase 2: out[l][d0..d0+63] = expT @ x ----------------
  {
    const int d0 = wave * 64;
    v8f o[4][4];
#pragma unroll
    for (int mt = 0; mt < 4; ++mt)
#pragma unroll
      for (int nt = 0; nt < 4; ++nt) o[mt][nt] = zero8;

#pragma unroll 1
    for (int k = 0; k < kS; k += 4) {
      const int kk = k + 2 * half;      // this lane's K pair
      v2f a[4], bf[4];
#pragma unroll
      for (int mt = 0; mt < 4; ++mt) {  // A = e^T : A[m=l][k=s] = e[s][l]
        const int m0 = 16 * mt + l16;
        v2f t;
        t.x = e[kk * kEP + m0];
        t.y = e[(kk + 1) * kEP + m0];
        a[mt] = t;
      }
#pragma unroll
      for (int nt = 0; nt < 4; ++nt) {  // B = x : B[k=s][n=d]
        const float* p = xb + (size_t)kk * kD + d0 + 16 * nt + l16;
        v2f t;
        t.x = p[0];
        t.y = p[kD];
        bf[nt] = t;
      }
#pragma unroll
      for (int mt = 0; mt < 4; ++mt)
#pragma unroll
        for (int nt = 0; nt < 4; ++nt)
          o[mt][nt] = wmma_f32_16x16x4(a[mt], bf[nt], o[mt][nt]);
    }

    // Normalize by 1/colsum (folded here) and store out[b, l0+m, d].
    float* ob = out + ((size_t)b * kL + l0) * kD;
#pragma unroll
    for (int mt = 0; mt < 4; ++mt)
#pragma unroll
      for (int nt = 0; nt < 4; ++nt) {
        const int dd = d0 + 16 * nt + l16;
#pragma unroll
        for (int i = 0; i < 8; ++i) {
          const int m = 16 * mt + i + 8 * half;
          ob[(size_t)m * kD + dd] = o[mt][nt][i] * colinv[m];
        }
      }
  }
}

extern "C" void kernel_launch(void* const* d_in, const int* in_sizes, int n_in,
                              void* d_out, int out_size, void* d_ws, size_t ws_size,
                              hipStream_t stream) {
  (void)in_sizes; (void)n_in; (void)out_size; (void)d_ws; (void)ws_size;
  const float*         x    = (const float*)d_in[0];
  const unsigned char* mask = (const unsigned char*)d_in[1];  // jax bool -> 1 byte
  const float*         lab  = (const float*)d_in[2];
  float*               out  = (float*)d_out;

  dim3 grid(kL / kLT, kB, 1);   // 128 label tiles x 16 batches = 2048 WGs
  dim3 block(256, 1, 1);        // 8 wave32
  size_t smem = (size_t)kSmemFloats * sizeof(float);  // 160768 B -> 2 WGs per 320KB WGP
  SoftmaxAttention_86749749444742_kernel<<<grid, block, smem, stream>>>(x, mask, lab, out);
}